// CannyFilterOpenCV_41085657153836
// MI455X (gfx1250) — compile-verified
//
#include <hip/hip_runtime.h>
#include <hip/hip_bf16.h>
#include <math.h>

// ---------------------------------------------------------------------------
// Canny (OpenCV-style) for x:[16,3,720,1280] f32 in [0,1] on gfx1250.
// HBM-bandwidth-bound stencil. gfx1250-specific paths used:
//   - global_load_async_to_lds_b32/b8 (ASYNCcnt) for halo-tile staging
//   - s_wait_asynccnt split counter waits
//   - wave32-sized tiles (16x16 = 8 waves, 32x24 = 24 waves)
// Intermediates live inside d_out (no d_ws dependency):
//   plane2 region (4*HW bytes) holds two uint8 state arrays (ping-pong).
// ---------------------------------------------------------------------------

#define B_  16
#define H_  720
#define W_  1280
#define HW_ (H_ * W_)

#define LOW_T  25.5f
#define HIGH_T 76.5f
#define PI_F   3.14159265358979323846f

#if defined(__AMDGCN__) && __has_builtin(__builtin_amdgcn_global_load_async_to_lds_b32) && __has_builtin(__builtin_amdgcn_s_wait_asynccnt)
#define USE_ASYNC_LDS 1
#else
#define USE_ASYNC_LDS 0
#endif

#if defined(__AMDGCN__) && __has_builtin(__builtin_amdgcn_global_load_async_to_lds_b8) && __has_builtin(__builtin_amdgcn_s_wait_asynccnt)
#define USE_ASYNC_LDS8 1
#else
#define USE_ASYNC_LDS8 0
#endif

typedef __attribute__((address_space(1))) int  glb_i32;
typedef __attribute__((address_space(3))) int  lds_i32;
typedef __attribute__((address_space(1))) char glb_i8;
typedef __attribute__((address_space(3))) char lds_i8;

__device__ __forceinline__ void async_ld32(const float* g, float* l) {
#if USE_ASYNC_LDS
  __builtin_amdgcn_global_load_async_to_lds_b32((glb_i32*)(g), (lds_i32*)(l), 0, 0);
#else
  *l = *g;
#endif
}

__device__ __forceinline__ void async_ld8(const unsigned char* g, unsigned char* l) {
#if USE_ASYNC_LDS8
  __builtin_amdgcn_global_load_async_to_lds_b8((glb_i8*)(g), (lds_i8*)(l), 0, 0);
#else
  *l = *g;
#endif
}

__device__ __forceinline__ void async_wait0() {
#if USE_ASYNC_LDS || USE_ASYNC_LDS8
  __builtin_amdgcn_s_wait_asynccnt(0);
#endif
}

// uint8 state arrays (ping-pong) live inside out's plane-2 region (4*HW bytes).
__device__ __forceinline__ unsigned char* state_base(float* out, int b) {
  return (unsigned char*)(out + ((size_t)b * 3 + 2) * HW_);
}

// ---------------------------------------------------------------------------
// Kernel 1 (fused): quantize + Sobel + channel argmax + bin + NMS + threshold.
// block 16x16 (8 wave32). Input halo 20x20x3 -> LDS (async b32).
// mag/bin computed for the 18x18 region in LDS; NMS fully in LDS.
// Writes state uint8 {0,1,2} into stateA (= plane2 bytes [0,HW)).
// ---------------------------------------------------------------------------
__global__ void __launch_bounds__(256)
canny_sobel_nms(const float* __restrict__ x, float* __restrict__ out) {
  __shared__ float         in_t[3 * 20 * 20];   // 4.8 KB
  __shared__ float         mag_t[18 * 18];      // 1.3 KB
  __shared__ unsigned char bin_t[18 * 18];

  const int b  = blockIdx.z;
  const int x0 = blockIdx.x * 16;
  const int y0 = blockIdx.y * 16;
  const int tx = threadIdx.x, ty = threadIdx.y;
  const int t  = ty * 16 + tx;

  const float* xb = x + (size_t)b * 3 * HW_;

  // stage 20x20x3 input halo (replicate border) via async copies
  for (int i = t; i < 3 * 400; i += 256) {
    const int ch  = i / 400;
    const int rem = i - ch * 400;
    const int r   = rem / 20;
    const int cc  = rem - r * 20;
    int gy = y0 - 2 + r;  gy = gy < 0 ? 0 : (gy > H_ - 1 ? H_ - 1 : gy);
    int gx = x0 - 2 + cc; gx = gx < 0 ? 0 : (gx > W_ - 1 ? W_ - 1 : gx);
    async_ld32(xb + (size_t)ch * HW_ + (size_t)gy * W_ + gx, &in_t[i]);
  }
  async_wait0();
  __syncthreads();

  // uint8 quantization in LDS
  for (int i = t; i < 3 * 400; i += 256) {
    float v = in_t[i];
    v = v < 0.f ? 0.f : (v > 1.f ? 1.f : v);
    in_t[i] = floorf(v * 255.f);
  }
  __syncthreads();

  // mag/bin for the 18x18 region: mag(r,c) is at global (y0-1+r, x0-1+c),
  // which sits at input-tile coords (r+1, c+1). OOB positions get mag=0
  // (matches the reference's zero-padded _shift in NMS).
  for (int j = t; j < 324; j += 256) {
    const int r = j / 18, c = j - (j / 18) * 18;
    const int gy = y0 - 1 + r, gx = x0 - 1 + c;
    if (gy < 0 || gy >= H_ || gx < 0 || gx >= W_) {
      mag_t[j] = 0.f;
      bin_t[j] = 0;
    } else {
      float best = -1.f, bgx = 0.f, bgy = 0.f;
#pragma unroll
      for (int ch = 0; ch < 3; ++ch) {
        const float* T = &in_t[ch * 400];
#define P_(rr, cc) T[(rr) * 20 + (cc)]
        const float gxv = (P_(r, c + 2) - P_(r, c)) +
                          2.f * (P_(r + 1, c + 2) - P_(r + 1, c)) +
                          (P_(r + 2, c + 2) - P_(r + 2, c));
        const float gyv = (P_(r + 2, c) + 2.f * P_(r + 2, c + 1) + P_(r + 2, c + 2)) -
                          (P_(r, c) + 2.f * P_(r, c + 1) + P_(r, c + 2));
#undef P_
        const float m = fabsf(gxv) + fabsf(gyv);
        if (m > best) { best = m; bgx = gxv; bgy = gyv; }  // argmax first-max rule
      }
      float ang = fmodf(atan2f(bgy, bgx), PI_F);
      if (ang < 0.f) ang += PI_F;  // [0, pi)
      mag_t[j] = best;
      bin_t[j] = (unsigned char)(((int)floorf((ang + PI_F / 8.f) / (PI_F / 4.f))) & 3);
    }
  }
  __syncthreads();

  // NMS + double threshold for own pixel
  static const int dy1[4] = {0, 1, 1, 1},    dx1[4] = {1, 1, 0, -1};
  static const int dy2[4] = {0, -1, -1, -1}, dx2[4] = {-1, -1, 0, 1};

  const int r = ty + 1, c = tx + 1;
  const float m  = mag_t[r * 18 + c];
  const int   bn = bin_t[r * 18 + c];
  const float n1 = mag_t[(r + dy1[bn]) * 18 + (c + dx1[bn])];
  const float n2 = mag_t[(r + dy2[bn]) * 18 + (c + dx2[bn])];
  const float nms = (m >= n1 && m >= n2) ? m : 0.f;
  const unsigned char s = nms > HIGH_T ? 2 : (nms > LOW_T ? 1 : 0);

  state_base(out, b)[(size_t)(y0 + ty) * W_ + (x0 + tx)] = s;  // stateA
}

// ---------------------------------------------------------------------------
// Kernel 2: one hysteresis pass on uint8 states — async-load 26x34 tile into
// LDS, monotone in-tile propagation (weak->strong if 8-adjacent strong) to
// local fixed point, write back interior. srcSel/dstSel in {0,1} select the
// ping-pong byte arrays inside plane2.
// block 32x24 = 768 threads = 24 wave32.
// ---------------------------------------------------------------------------
#define TX_ 32
#define TY_ 24

__global__ void __launch_bounds__(768)
canny_hyst(float* __restrict__ out, int srcSel, int dstSel) {
  __shared__ unsigned char st[(TY_ + 2) * (TX_ + 2)];  // 26*34 = 884 B
  __shared__ int changed;

  const int b  = blockIdx.z;
  const int x0 = blockIdx.x * TX_;
  const int y0 = blockIdx.y * TY_;
  const int tx = threadIdx.x, ty = threadIdx.y;
  const int t  = ty * TX_ + tx;

  unsigned char* base = state_base(out, b);
  const unsigned char* sp = base + (size_t)srcSel * HW_;
  unsigned char*       dp = base + (size_t)dstSel * HW_;

  for (int i = t; i < 26 * 34; i += 768) {
    const int r = i / 34, c = i - (i / 34) * 34;
    const int gy = y0 - 1 + r, gx = x0 - 1 + c;
    if (gy >= 0 && gy < H_ && gx >= 0 && gx < W_)
      async_ld8(sp + (size_t)gy * W_ + gx, &st[i]);
    else
      st[i] = 0;  // zero-padded border (reduce_window 'SAME', init 0)
  }
  async_wait0();
  __syncthreads();

  const int r = ty + 1, c = tx + 1;
  for (int it = 0; it < 64; ++it) {
    if (t == 0) changed = 0;
    __syncthreads();
    if (st[r * 34 + c] == 1) {
      const bool any =
          st[(r - 1) * 34 + (c - 1)] == 2 || st[(r - 1) * 34 + c] == 2 ||
          st[(r - 1) * 34 + (c + 1)] == 2 || st[r * 34 + (c - 1)] == 2 ||
          st[r * 34 + (c + 1)] == 2       || st[(r + 1) * 34 + (c - 1)] == 2 ||
          st[(r + 1) * 34 + c] == 2       || st[(r + 1) * 34 + (c + 1)] == 2;
      if (any) { st[r * 34 + c] = 2; changed = 1; }  // monotone 1->2; race benign
    }
    __syncthreads();
    if (!changed) break;
  }

  dp[(size_t)(y0 + ty) * W_ + (x0 + tx)] = st[r * 34 + c];
}

// ---------------------------------------------------------------------------
// Kernel 3: expand final byte state (stateA, sel 0) -> plane0 floats {0,1}.
// Reads plane2-region bytes, writes plane0 floats: disjoint, no hazard.
// ---------------------------------------------------------------------------
__global__ void __launch_bounds__(256)
canny_expand(float* __restrict__ out) {
  const size_t i = (size_t)blockIdx.x * 256 + threadIdx.x;
  if (i >= (size_t)B_ * HW_) return;
  const size_t b = i / HW_;
  const size_t p = i - b * HW_;
  const unsigned char* sa = state_base(out, (int)b);
  out[b * 3 * HW_ + p] = (sa[p] == (unsigned char)2) ? 1.f : 0.f;
}

// ---------------------------------------------------------------------------
// Kernel 4: broadcast plane0 -> planes 1 and 2.
// Reads plane0 only, writes planes 1/2 only: disjoint, no hazard.
// ---------------------------------------------------------------------------
__global__ void __launch_bounds__(256)
canny_broadcast(float* __restrict__ out) {
  const size_t i = (size_t)blockIdx.x * 256 + threadIdx.x;
  if (i >= (size_t)B_ * HW_) return;
  const size_t b = i / HW_;
  const size_t p = i - b * HW_;
  float* ob = out + b * 3 * HW_;
  const float v = ob[p];
  ob[HW_ + p]     = v;
  ob[2 * HW_ + p] = v;
}

// ---------------------------------------------------------------------------
extern "C" void kernel_launch(void* const* d_in, const int* in_sizes, int n_in,
                              void* d_out, int out_size, void* d_ws, size_t ws_size,
                              hipStream_t stream) {
  (void)in_sizes; (void)n_in; (void)out_size; (void)d_ws; (void)ws_size;
  const float* x = (const float*)d_in[0];
  float* out = (float*)d_out;

  const size_t nPix = (size_t)B_ * HW_;
  const int nBlk1D = (int)((nPix + 255) / 256);

  // fused quantize + Sobel + argmax + NMS + threshold -> stateA bytes
  canny_sobel_nms<<<dim3(W_ / 16, H_ / 16, B_), dim3(16, 16), 0, stream>>>(x, out);

  // 10 ping-pong hysteresis passes (even count -> final state back in stateA).
  // Each pass converges in-tile (32x24) and crosses one tile boundary:
  // ~270 px propagation radius.
  for (int i = 0; i < 10; ++i) {
    const int s = i & 1;  // 0: A->B, 1: B->A
    canny_hyst<<<dim3(W_ / TX_, H_ / TY_, B_), dim3(TX_, TY_), 0, stream>>>(out, s, 1 - s);
  }

  canny_expand<<<nBlk1D, 256, 0, stream>>>(out);
  canny_broadcast<<<nBlk1D, 256, 0, stream>>>(out);
}